// DeepReservoir_15333033247304
// MI455X (gfx1250) — compile-verified
//
#include <hip/hip_runtime.h>
#include <math.h>

typedef __attribute__((ext_vector_type(2))) float v2f;
typedef __attribute__((ext_vector_type(8))) float v8f;

#define NWG   32          // 2 batch groups x 16 N-slice workgroups
#define BLK   256         // 8 wave32s
#define TT    1024
#define II    64
#define UU    512
#define ALPHA 0.9f

// ---- dynamic LDS layout (floats) ----
// xw   [64][32]  pair-packed   @ 0       (2048)
// r0l  [512][32] pair-packed   @ 2048    (16384)
// w1l  [512][32] pair-packed   @ 18432   (16384)
// r1l  [512][32] pair-packed   @ 34816   (16384)
// h0l  [16][512]               @ 51200   (8192)
// h1l  [16][512]               @ 59392   (8192)
// xt   [16][64]                @ 67584   (1024)
// scr  [8][256]                @ 68608   (2048)
// b0s  [32]                    @ 70656
// b1s  [32]                    @ 70688
#define LDS_FLOATS 70720
#define LDS_BYTES  (LDS_FLOATS * 4)

__global__ void init_sync_kernel(int* sync) {
    if (threadIdx.x == 0) { sync[0] = 0; sync[1] = 0; }
}

__device__ __forceinline__ void grid_barrier(int* cnt, int* gen) {
    __syncthreads();
    if (threadIdx.x == 0) {
        __threadfence();
        int g = __hip_atomic_load(gen, __ATOMIC_RELAXED, __HIP_MEMORY_SCOPE_AGENT);
        int a = __hip_atomic_fetch_add(cnt, 1, __ATOMIC_ACQ_REL, __HIP_MEMORY_SCOPE_AGENT);
        if (a == NWG - 1) {
            __hip_atomic_store(cnt, 0, __ATOMIC_RELAXED, __HIP_MEMORY_SCOPE_AGENT);
            __hip_atomic_fetch_add(gen, 1, __ATOMIC_ACQ_REL, __HIP_MEMORY_SCOPE_AGENT);
        } else {
            while (__hip_atomic_load(gen, __ATOMIC_ACQUIRE, __HIP_MEMORY_SCOPE_AGENT) == g) {
                __builtin_amdgcn_s_sleep(2);
            }
        }
    }
    __syncthreads();
    __threadfence();  // make peer-WG h writes visible to all threads' loads
}

// C += A[16 x (k1-k0)] * W[(k1-k0) x 16] with v_wmma_f32_16x16x4_f32.
// A: row-major f32, leading dim lda (8B-aligned rows, k even).
// Wp: pair-packed weight slice: Wp[(k>>1)*64 + col*2 + (k&1)].
// Both operand fragments are a single contiguous v2f (ds_load_b64) per WMMA:
//   A pair  = Arow[(kb>>1)+kh], B pair = Wp[((kb>>1)+kh)*64 + (nb+mn)*2].
// Dual accumulators break the C dependency chain. k0,k1 multiples of 4, wave-uniform.
__device__ __forceinline__ void wmma_accum(v8f& ca, v8f& cb,
    const float* A, int lda, const float* Wp, int nb,
    int k0, int k1, int mn, int kh)
{
    const v2f* ap = (const v2f*)(A + mn * lda);       // indexed by k-pair
    const v2f* bp = (const v2f*)(Wp + (nb + mn) * 2); // + kpair*32 v2f's (64 floats)
    int kb = k0;
    for (; kb + 8 <= k1; kb += 8) {
        {
            int kp = (kb >> 1) + kh;
            v2f av = ap[kp];
            v2f bv = bp[kp * 32];
            ca = __builtin_amdgcn_wmma_f32_16x16x4_f32(false, av, false, bv, (short)0, ca, false, false);
        }
        {
            int kp = (kb >> 1) + 2 + kh;
            v2f av = ap[kp];
            v2f bv = bp[kp * 32];
            cb = __builtin_amdgcn_wmma_f32_16x16x4_f32(false, av, false, bv, (short)0, cb, false, false);
        }
    }
    for (; kb < k1; kb += 4) {
        int kp = (kb >> 1) + kh;
        v2f av = ap[kp];
        v2f bv = bp[kp * 32];
        ca = __builtin_amdgcn_wmma_f32_16x16x4_f32(false, av, false, bv, (short)0, ca, false, false);
    }
}

__launch_bounds__(BLK)
__global__ void reservoir_kernel(const float* __restrict__ x,
                                 const float* __restrict__ Win0, const float* __restrict__ Rec0,
                                 const float* __restrict__ B0,
                                 const float* __restrict__ Win1, const float* __restrict__ Rec1,
                                 const float* __restrict__ B1,
                                 float* __restrict__ out,
                                 float* __restrict__ hbuf, int* __restrict__ sync)
{
    extern __shared__ float lds[];
    float* xw  = lds;             // [32][32][2]  (64 rows pair-packed)
    float* r0l = lds + 2048;      // [256][32][2]
    float* w1l = lds + 18432;     // [256][32][2]
    float* r1l = lds + 34816;     // [256][32][2]
    float* h0l = lds + 51200;     // [16][512]
    float* h1l = lds + 59392;     // [16][512]
    float* xt  = lds + 67584;     // [16][64]
    float* scr = lds + 68608;     // [8][256]
    float* b0s = lds + 70656;     // [32]
    float* b1s = lds + 70688;     // [32]

    const int wg  = blockIdx.x;
    const int g   = wg >> 4;          // batch group (rows g*16..g*16+15)
    const int c0  = (wg & 15) * 32;   // this WG's output-column base
    const int tid = threadIdx.x;

    // ---- preload weight slices into pair-packed LDS (once) ----
    for (int idx = tid; idx < 64 * 32; idx += BLK) {
        int k = idx >> 5, c = idx & 31;
        int p = (k >> 1) * 64 + c * 2 + (k & 1);
        xw[p] = Win0[k * UU + c0 + c];
    }
    for (int idx = tid; idx < 512 * 32; idx += BLK) {
        int k = idx >> 5, c = idx & 31;
        int p = (k >> 1) * 64 + c * 2 + (k & 1);
        r0l[p] = Rec0[k * UU + c0 + c];
        w1l[p] = Win1[k * UU + c0 + c];
        r1l[p] = Rec1[k * UU + c0 + c];
    }
    if (tid < 32) { b0s[tid] = B0[c0 + tid]; b1s[tid] = B1[c0 + tid]; }
    for (int idx = tid; idx < 16 * UU; idx += BLK) { h0l[idx] = 0.f; h1l[idx] = 0.f; }
    __syncthreads();

    const int wv = tid >> 5;          // wave 0..7
    const int lid = tid & 31;
    const int nt = wv & 1;            // n-tile (0/1)
    const int kq = wv >> 1;           // k-quarter (0..3)
    const int mn = lid & 15;
    const int kh = lid >> 4;
    const int nb = nt * 16;

    int* cnt = sync;
    int* gen = sync + 1;
    float* h0g = hbuf + (size_t)(g * 2 + 0) * (16 * UU);
    float* h1g = hbuf + (size_t)(g * 2 + 1) * (16 * UU);
    float* sw  = scr + wv * 256;

    for (int t = 0; t < TT; ++t) {
        // stage x_t [16][64] into LDS: one float4 per thread
        {
            int m = tid >> 4, i4 = tid & 15;
            const float4* src = (const float4*)(x + ((size_t)(g * 16 + m) * TT + t) * II);
            ((float4*)xt)[m * 16 + i4] = src[i4];
        }
        __syncthreads();

        // ===== layer 0: pre = x_t @ Win0 + h0 @ Rec0  (combined K = 576, split 4 ways) =====
        {
            v8f ca = {}; v8f cb = {};
            int kc0 = kq * 144, kc1 = kc0 + 144;
            if (kc0 < II) {                                  // x segment [0,64)
                int xe = (kc1 < II) ? kc1 : II;
                wmma_accum(ca, cb, xt, II, xw, nb, kc0, xe, mn, kh);
            }
            int hs = (kc0 > II ? kc0 : II) - II;             // h segment
            int he = kc1 - II;
            wmma_accum(ca, cb, h0l, UU, r0l, nb, hs, he, mn, kh);
            for (int r = 0; r < 8; ++r) sw[r * 32 + lid] = ca[r] + cb[r];
        }
        __syncthreads();
        if (wv < 2) {   // waves 0/1 reduce 4 k-partials for their n-tile, finalize
            for (int r = 0; r < 8; ++r) {
                int e = r * 32 + lid;
                float s = scr[(wv + 0) * 256 + e] + scr[(wv + 2) * 256 + e]
                        + scr[(wv + 4) * 256 + e] + scr[(wv + 6) * 256 + e];
                int nl = wv * 16 + mn;          // local col 0..31 (wv == nt here)
                int m  = r + 8 * kh;            // C layout: M = r + 8*(lane/16)
                float pre  = s + b0s[nl];
                float hnew = (1.0f - ALPHA) * h0l[m * UU + c0 + nl] + ALPHA * tanhf(pre);
                h0g[m * UU + c0 + nl] = hnew;
                out[((size_t)(g * 16 + m) * TT + t) * 1024 + c0 + nl] = hnew;
            }
            __threadfence();
        }
        grid_barrier(cnt, gen);
        for (int idx = tid; idx < 16 * UU; idx += BLK) h0l[idx] = h0g[idx];  // h0 <- s0_t (full)
        __syncthreads();

        // ===== layer 1: pre = s0_t @ Win1 + h1 @ Rec1  (combined K = 1024, split 4 ways) =====
        {
            v8f ca = {}; v8f cb = {};
            if (kq < 2) wmma_accum(ca, cb, h0l, UU, w1l, nb, kq * 256, kq * 256 + 256, mn, kh);
            else        wmma_accum(ca, cb, h1l, UU, r1l, nb, (kq - 2) * 256, (kq - 2) * 256 + 256, mn, kh);
            for (int r = 0; r < 8; ++r) sw[r * 32 + lid] = ca[r] + cb[r];
        }
        __syncthreads();
        if (wv < 2) {
            for (int r = 0; r < 8; ++r) {
                int e = r * 32 + lid;
                float s = scr[(wv + 0) * 256 + e] + scr[(wv + 2) * 256 + e]
                        + scr[(wv + 4) * 256 + e] + scr[(wv + 6) * 256 + e];
                int nl = wv * 16 + mn;
                int m  = r + 8 * kh;
                float pre  = s + b1s[nl];
                float hnew = (1.0f - ALPHA) * h1l[m * UU + c0 + nl] + ALPHA * tanhf(pre);
                h1g[m * UU + c0 + nl] = hnew;
                out[((size_t)(g * 16 + m) * TT + t) * 1024 + 512 + c0 + nl] = hnew;
            }
            __threadfence();
        }
        grid_barrier(cnt, gen);
        for (int idx = tid; idx < 16 * UU; idx += BLK) h1l[idx] = h1g[idx];
        __syncthreads();
    }
}

extern "C" void kernel_launch(void* const* d_in, const int* in_sizes, int n_in,
                              void* d_out, int out_size, void* d_ws, size_t ws_size,
                              hipStream_t stream) {
    (void)in_sizes; (void)n_in; (void)out_size; (void)ws_size;
    const float* x  = (const float*)d_in[0];
    const float* k0 = (const float*)d_in[1];
    const float* r0 = (const float*)d_in[2];
    const float* b0 = (const float*)d_in[3];
    const float* k1 = (const float*)d_in[4];
    const float* r1 = (const float*)d_in[5];
    const float* b1 = (const float*)d_in[6];
    float* out = (float*)d_out;

    int*   sync = (int*)d_ws;
    float* hbuf = (float*)((char*)d_ws + 256);   // 4 x 32KB h-exchange buffers

    (void)hipFuncSetAttribute(reinterpret_cast<const void*>(reservoir_kernel),
                              hipFuncAttributeMaxDynamicSharedMemorySize, LDS_BYTES);

    init_sync_kernel<<<dim3(1), dim3(32), 0, stream>>>(sync);
    reservoir_kernel<<<dim3(NWG), dim3(BLK), LDS_BYTES, stream>>>(
        x, k0, r0, b0, k1, r1, b1, out, hbuf, sync);
}